// LocalTrittention_85804856639905
// MI455X (gfx1250) — compile-verified
//
#include <hip/hip_runtime.h>

typedef float v2f __attribute__((ext_vector_type(2)));
typedef float v8f __attribute__((ext_vector_type(8)));

#define BS 2
#define TS 4096
#define DM 512
#define NH 8
#define DH 64
#define WIN 16
#define NW (TS / WIN)        // 256 windows
#define NC (5 * NH * DH)     // 2560 columns of abcde
#define MROWS (BS * TS)      // 8192
#define IGNORE_VAL (-1000000.0f)

// ---------------------------------------------------------------------------
// f32 WMMA wrapper: D(16x16) += A(16x4) * B(4x16), all fp32, wave32.
// A frag: lane L holds row M=L%16, K = 2*(L>=16) + r  (r = component)
// B frag: lane L holds col N=L%16, K = 2*(L>=16) + r
// C/D   : lane L, component v -> row M = v + 8*(L>=16), col N = L%16
// ---------------------------------------------------------------------------
__device__ __forceinline__ v8f wmma4(v2f a, v2f b, v8f c) {
  return __builtin_amdgcn_wmma_f32_16x16x4_f32(
      /*neg_a=*/false, a, /*neg_b=*/false, b,
      /*c_mod=*/(short)0, c, /*reuse_a=*/false, /*reuse_b=*/false);
}

// CDNA5 async global->LDS copy (VGLOBAL, tracked with ASYNCcnt).
// LDS byte address = low 32 bits of the generic pointer (LDS aperture maps
// addr[31:0] directly). 16B global & LDS alignment required.
__device__ __forceinline__ void async_ld_b128(const float* gptr, float* lptr) {
  const unsigned lds = (unsigned)(unsigned long long)(uintptr_t)lptr;
  asm volatile("global_load_async_to_lds_b128 %0, %1, off"
               :
               : "v"(lds), "v"(gptr)
               : "memory");
}
__device__ __forceinline__ void wait_async0() {
  asm volatile("s_wait_asynccnt 0x0" ::: "memory");
}

// ---------------------------------------------------------------------------
// C[M,N] = A[M,K] @ B[K,N] + bias[N]   (fp32, row-major, M%256==N%128==K%16==0)
// 256 threads = 8 waves; block tile 256x128; wave tile 64x64 (4x4 WMMA tiles).
// Double-buffered LDS filled with global_load_async_to_lds_b128:
//   wait_async(0) -> barrier -> issue chunk i+1 -> compute chunk i.
// ---------------------------------------------------------------------------
__global__ __launch_bounds__(256) void gemm_bias_f32(
    const float* __restrict__ A, const float* __restrict__ B,
    const float* __restrict__ bias, float* __restrict__ C,
    int M, int N, int K) {
  __shared__ float As[2][256 * 20];   // 256 rows x 16 K, stride 20 (bank-safe, 16B-align)
  __shared__ float Bs[2][16 * 136];   // 16 K x 128 cols, stride 136 (2*136%64==16)

  const int tid  = threadIdx.x;
  const int lane = tid & 31;
  const int wid  = tid >> 5;
  const int n    = lane & 15;
  const int hi   = lane >> 4;
  const int wm   = wid >> 1;       // 0..3  -> 64-row slab
  const int wn   = wid & 1;        // 0..1  -> 64-col slab
  const int gm0  = blockIdx.y * 256;
  const int gn0  = blockIdx.x * 128;

  const v8f vzero = {0.f, 0.f, 0.f, 0.f, 0.f, 0.f, 0.f, 0.f};
  v8f acc[4][4];
#pragma unroll
  for (int tm = 0; tm < 4; ++tm)
#pragma unroll
    for (int tn = 0; tn < 4; ++tn) acc[tm][tn] = vzero;

  // A loader: 4 x b128 per thread: row = i*64 + (tid>>2), cols (tid&3)*4..+3
  const int ar = tid >> 2;              // 0..63 (base row within 64-row group)
  const int ac = (tid & 3) * 4;         // 0,4,8,12
  // B loader: 2 x b128 per thread: row = i*8 + (tid>>5), col = (tid&31)*4
  const int br = tid >> 5;              // 0..7
  const int bc = (tid & 31) * 4;        // 0..124

  const int nch = K / 16;

  // ---- prologue: async-load chunk 0 into buffer 0 ----
  {
    const float* Abase = A + (size_t)(gm0)*K;        // + row*K + kc + col
#pragma unroll
    for (int i = 0; i < 4; ++i) {
      const int row = i * 64 + ar;
      async_ld_b128(Abase + (size_t)row * K + ac, &As[0][row * 20 + ac]);
    }
#pragma unroll
    for (int i = 0; i < 2; ++i) {
      const int row = i * 8 + br;
      async_ld_b128(B + (size_t)row * N + gn0 + bc, &Bs[0][row * 136 + bc]);
    }
  }

  for (int ch = 0; ch < nch; ++ch) {
    const int cur = ch & 1;
    wait_async0();        // this wave's pending chunk has landed in LDS
    __syncthreads();      // everyone's chunk landed; prev buffer reads done

    if (ch + 1 < nch) {   // overlap next chunk's DMA with this chunk's WMMAs
      const int kc = (ch + 1) * 16;
      const int nxt = cur ^ 1;
      const float* Abase = A + (size_t)(gm0)*K + kc;
#pragma unroll
      for (int i = 0; i < 4; ++i) {
        const int row = i * 64 + ar;
        async_ld_b128(Abase + (size_t)row * K + ac, &As[nxt][row * 20 + ac]);
      }
#pragma unroll
      for (int i = 0; i < 2; ++i) {
        const int row = i * 8 + br;
        async_ld_b128(B + (size_t)(kc + row) * N + gn0 + bc,
                      &Bs[nxt][row * 136 + bc]);
      }
    }

#pragma unroll
    for (int kk = 0; kk < 4; ++kk) {
      const int K0 = kk * 4 + 2 * hi;
      v2f av[4], bv[4];
#pragma unroll
      for (int tm = 0; tm < 4; ++tm)
        av[tm] = *(const v2f*)&As[cur][(wm * 64 + tm * 16 + n) * 20 + K0];
#pragma unroll
      for (int tn = 0; tn < 4; ++tn) {
        const int cb = wn * 64 + tn * 16 + n;
        bv[tn].x = Bs[cur][(K0 + 0) * 136 + cb];
        bv[tn].y = Bs[cur][(K0 + 1) * 136 + cb];
      }
#pragma unroll
      for (int tm = 0; tm < 4; ++tm)
#pragma unroll
        for (int tn = 0; tn < 4; ++tn)
          acc[tm][tn] = wmma4(av[tm], bv[tn], acc[tm][tn]);
    }
  }

  // ---- epilogue: bias + store ----
#pragma unroll
  for (int tn = 0; tn < 4; ++tn) {
    const int cn = gn0 + wn * 64 + tn * 16 + n;
    const float bb = bias[cn];
#pragma unroll
    for (int tm = 0; tm < 4; ++tm) {
#pragma unroll
      for (int v = 0; v < 8; ++v) {
        const int r = gm0 + wm * 64 + tm * 16 + v + 8 * hi;
        C[(size_t)r * N + cn] = acc[tm][tn][v] + bb;
      }
    }
  }
}

// ---------------------------------------------------------------------------
// Windowed trittention. One block per (zn = bsi*8+h, nw). 256 threads = 8 waves.
// Scores: attn(16x1024) = c(16x64) @ P^T, P[(m,l),d] = a_la[m,d]*b_la[l,d].
// Wave wid owns N-tiles j = wid*8 .. wid*8+7 (m = j>>1, l = (j&1)*16 + n).
// Softmax in registers (half-wave shuffles + LDS cross-wave stage), then
// z(16x64) = [rowsum(s) | colsum(s)](16x64) @ [d_la; e_la](64x64) via WMMA.
// ---------------------------------------------------------------------------
__global__ __launch_bounds__(256) void trittention_win(
    const float* __restrict__ P, float* __restrict__ Z) {
  __shared__ float cS[16 * 68];
  __shared__ float aS[32 * 68];
  __shared__ float bS[32 * 68];
  __shared__ float dS[32 * 68];
  __shared__ float eS[32 * 68];
  __shared__ float U[16 * 68];     // cols 0..31 = sm, 32..63 = sl
  __shared__ float pmax[16 * 8];
  __shared__ float psum[16 * 8];

  const int nw  = blockIdx.x;
  const int zn  = blockIdx.y;
  const int bsi = zn >> 3;
  const int h   = zn & 7;

  const int tid  = threadIdx.x;
  const int lane = tid & 31;
  const int wid  = tid >> 5;
  const int n    = lane & 15;
  const int hi   = lane >> 4;

  for (int i = tid; i < 16 * 68; i += 256) U[i] = 0.0f;

  // ---- Load c (current window) and a/b/d/e look-around tiles into LDS ----
  {
    const int cr = tid >> 4;                  // 0..15
    const int cc = (tid & 15) * 4;            // 0..60
    const size_t crow =
        (size_t)(bsi * TS + nw * WIN + cr) * NC + 2 * DM + h * DH + cc;
    *(float4*)&cS[cr * 68 + cc] = *(const float4*)(P + crow);

    const int r  = tid >> 3;                  // 0..31 look-around row
    const int c0 = (tid & 7) * 8;             // 0..56
    const bool valid = (r >= 16) || (nw > 0);
    const int tok = (r < 16) ? ((nw - 1) * WIN + r) : (nw * WIN + (r - 16));
    const float4 z4 = make_float4(0.f, 0.f, 0.f, 0.f);
    float4 va0 = z4, va1 = z4, vb0 = z4, vb1 = z4;
    float4 vd0 = z4, vd1 = z4, ve0 = z4, ve1 = z4;
    if (valid) {
      const float* p0 = P + (size_t)(bsi * TS + tok) * NC + h * DH + c0;
      va0 = *(const float4*)(p0 + 0 * DM); va1 = *(const float4*)(p0 + 0 * DM + 4);
      vb0 = *(const float4*)(p0 + 1 * DM); vb1 = *(const float4*)(p0 + 1 * DM + 4);
      vd0 = *(const float4*)(p0 + 3 * DM); vd1 = *(const float4*)(p0 + 3 * DM + 4);
      ve0 = *(const float4*)(p0 + 4 * DM); ve1 = *(const float4*)(p0 + 4 * DM + 4);
    }
    *(float4*)&aS[r * 68 + c0] = va0; *(float4*)&aS[r * 68 + c0 + 4] = va1;
    *(float4*)&bS[r * 68 + c0] = vb0; *(float4*)&bS[r * 68 + c0 + 4] = vb1;
    *(float4*)&dS[r * 68 + c0] = vd0; *(float4*)&dS[r * 68 + c0 + 4] = vd1;
    *(float4*)&eS[r * 68 + c0] = ve0; *(float4*)&eS[r * 68 + c0 + 4] = ve1;
  }
  __syncthreads();

  // ---- Phase 1: scores via WMMA, 8 tiles/wave, K=64 ----
  const v8f vzero = {0.f, 0.f, 0.f, 0.f, 0.f, 0.f, 0.f, 0.f};
  v8f acc[8];
#pragma unroll
  for (int i = 0; i < 8; ++i) acc[i] = vzero;

  for (int kk = 0; kk < 16; ++kk) {
    const int K0 = kk * 4 + 2 * hi;
    v2f av = *(const v2f*)&cS[n * 68 + K0];
#pragma unroll
    for (int i = 0; i < 8; ++i) {
      const int j = wid * 8 + i;
      const int m = j >> 1;
      const int l = (j & 1) * 16 + n;
      v2f bv;
      bv.x = aS[m * 68 + K0]     * bS[l * 68 + K0];
      bv.y = aS[m * 68 + K0 + 1] * bS[l * 68 + K0 + 1];
      acc[i] = wmma4(av, bv, acc[i]);
    }
  }

  // ---- Mask + scale (replicates where(mask | attn==0, IGNORE, attn)/dh) ----
#pragma unroll
  for (int i = 0; i < 8; ++i) {
    const int j = wid * 8 + i;
    const int m = j >> 1;
    const int l = (j & 1) * 16 + n;
    const int mpos = (m < 16) ? ((nw == 0) ? 0 : ((nw - 1) * WIN + m))
                              : (nw * WIN + (m - 16));
    const int lpos = (l < 16) ? ((nw == 0) ? 0 : ((nw - 1) * WIN + l))
                              : (nw * WIN + (l - 16));
#pragma unroll
    for (int v = 0; v < 8; ++v) {
      const int qpos = nw * WIN + v + 8 * hi;
      const float x = acc[i][v];
      const bool msk = (qpos < lpos) || (lpos <= mpos) || (x == 0.0f);
      acc[i][v] = (msk ? IGNORE_VAL : x) * (1.0f / 64.0f);
    }
  }

  // ---- Softmax over 1024 entries per row w (= v + 8*hi) ----
  float rmax[8], rinv[8];
#pragma unroll
  for (int v = 0; v < 8; ++v) {
    float mx = acc[0][v];
#pragma unroll
    for (int i = 1; i < 8; ++i) mx = fmaxf(mx, acc[i][v]);
#pragma unroll
    for (int off = 1; off < 16; off <<= 1)
      mx = fmaxf(mx, __shfl_xor(mx, off, 32));
    if (n == 0) pmax[(v + 8 * hi) * 8 + wid] = mx;
  }
  __syncthreads();
#pragma unroll
  for (int v = 0; v < 8; ++v) {
    float mx = pmax[(v + 8 * hi) * 8];
#pragma unroll
    for (int k = 1; k < 8; ++k) mx = fmaxf(mx, pmax[(v + 8 * hi) * 8 + k]);
    rmax[v] = mx;
  }
#pragma unroll
  for (int v = 0; v < 8; ++v) {
    float s = 0.f;
#pragma unroll
    for (int i = 0; i < 8; ++i) {
      const float e = __expf(acc[i][v] - rmax[v]);
      acc[i][v] = e;
      s += e;
    }
#pragma unroll
    for (int off = 1; off < 16; off <<= 1) s += __shfl_xor(s, off, 32);
    if (n == 0) psum[(v + 8 * hi) * 8 + wid] = s;
  }
  __syncthreads();
#pragma unroll
  for (int v = 0; v < 8; ++v) {
    float s = 0.f;
#pragma unroll
    for (int k = 0; k < 8; ++k) s += psum[(v + 8 * hi) * 8 + k];
    rinv[v] = 1.0f / s;
  }

  // ---- sm[w][m] (plain stores) and sl[w][l] (ds_add_f32 atomics) into U ----
#pragma unroll
  for (int v = 0; v < 8; ++v) {
    const int w = v + 8 * hi;
    float sl0 = 0.f, sl1 = 0.f;
#pragma unroll
    for (int p = 0; p < 4; ++p) {
      const float s0 = acc[2 * p][v] * rinv[v];
      const float s1 = acc[2 * p + 1][v] * rinv[v];
      float t = s0 + s1;
#pragma unroll
      for (int off = 1; off < 16; off <<= 1) t += __shfl_xor(t, off, 32);
      if (n == 0) U[w * 68 + (4 * wid + p)] = t;   // m = 4*wid + p
      sl0 += s0;
      sl1 += s1;
    }
    atomicAdd(&U[w * 68 + 32 + n], sl0);           // l = n
    atomicAdd(&U[w * 68 + 48 + n], sl1);           // l = 16 + n
  }
  __syncthreads();

  // ---- Phase 3: z(16x64) = U(16x64) @ [d_la; e_la](64x64), waves 0..3 ----
  if (wid < 4) {
    v8f zacc = vzero;
    const int col = wid * 16 + n;
    for (int kk = 0; kk < 16; ++kk) {
      const int K0 = kk * 4 + 2 * hi;
      v2f av = *(const v2f*)&U[n * 68 + K0];
      v2f bv;
      bv.x = (K0 < 32) ? dS[K0 * 68 + col] : eS[(K0 - 32) * 68 + col];
      bv.y = (K0 + 1 < 32) ? dS[(K0 + 1) * 68 + col] : eS[(K0 - 31) * 68 + col];
      zacc = wmma4(av, bv, zacc);
    }
#pragma unroll
    for (int v = 0; v < 8; ++v) {
      const int t = nw * WIN + v + 8 * hi;
      Z[(size_t)(bsi * TS + t) * DM + h * DH + col] = zacc[v];
    }
  }
}

// ---------------------------------------------------------------------------
extern "C" void kernel_launch(void* const* d_in, const int* in_sizes, int n_in,
                              void* d_out, int out_size, void* d_ws,
                              size_t ws_size, hipStream_t stream) {
  (void)in_sizes; (void)n_in; (void)out_size; (void)ws_size;
  const float* x   = (const float*)d_in[0];   // (2,4096,512)
  const float* Wab = (const float*)d_in[1];   // (512,2560)
  const float* bab = (const float*)d_in[2];   // (2560,)
  const float* WO  = (const float*)d_in[3];   // (512,512)
  const float* bO  = (const float*)d_in[4];   // (512,)

  float* out = (float*)d_out;                       // output 0: (2,4096,512)
  float* z   = out + (size_t)BS * TS * DM;          // output 1: (2,4096,512)
  float* abcde = (float*)d_ws;                      // (8192, 2560) fp32

  // 1) abcde = x @ W_abcde + b_abcde
  dim3 g1(NC / 128, MROWS / 256);
  gemm_bias_f32<<<g1, 256, 0, stream>>>(x, Wab, bab, abcde, MROWS, NC, DM);

  // 2) windowed trittention -> z (written directly into d_out's second half)
  dim3 g2(NW, BS * NH);
  trittention_win<<<g2, 256, 0, stream>>>(abcde, z);

  // 3) out = z @ W_O + b_O
  dim3 g3(DM / 128, MROWS / 256);
  gemm_bias_f32<<<g3, 256, 0, stream>>>(z, WO, bO, out, MROWS, DM, DM);
}